// ContiguousMatch_15822659518933
// MI455X (gfx1250) — compile-verified
//
#include <hip/hip_runtime.h>

// ContiguousMatch CDR3 scoring on gfx1250 (MI455X).
// v_match via V_WMMA_F32_16X16X4_F32 with germline as the A-matrix
// (M = sequence position) and x as the B-matrix (N = batch element):
// D lands positions-in-VGPRs / batch-in-lanes, so the cumprod over
// positions is lane-local. j_score is an FMA reduction co-executing
// with the WMMA chain. V germline rows are preloaded per-lane per tile
// so the inner loop has no V LDS traffic and no divergent guards.

typedef __attribute__((ext_vector_type(2))) float v2f;
typedef __attribute__((ext_vector_type(8))) float v8f;

#define SEQ_L 40
#define ALPH  21
#define ROWE  (SEQ_L * ALPH)   // 840 floats per batch row

__device__ __forceinline__ v8f wmma4(v2f a, v2f b, v8f c) {
  // D = A(16x4 f32) * B(4x16 f32) + C(16x16 f32)
  return __builtin_amdgcn_wmma_f32_16x16x4_f32(
      /*neg_a=*/false, a, /*neg_b=*/false, b,
      /*c_mod=*/(short)0, c, /*reuse_a=*/false, /*reuse_b=*/false);
}

struct TileCtx {
  const float* xrow;   // this lane's batch row
  const float* sV;
  const float* sJ;
  int half;            // lane>>4 : which K-pair half
  int n;               // lane&15 : batch column == A row id
};

// One 16-position tile: for l in [l0, l0+cnt), accumulate v_match rows
// (row = l-l0) into acc and J dot partials into jacc0/jacc1.
__device__ __forceinline__ void tile_mma(const TileCtx& t, int l0, int cnt,
                                         v8f& acc, float& jacc0, float& jacc1) {
  // ---- preload this lane's V germline row for the whole tile ----
  // Lane n is the active A row exactly when li == n, i.e. l == l0+n.
  int myl = l0 + t.n; if (myl > SEQ_L - 1) myl = SEQ_L - 1;   // clamp (unused lanes)
  const int vb = myl * ALPH;
  float va0[6], va1[6];
#pragma unroll
  for (int c = 0; c < 6; ++c) {
    const int a0 = 4 * c + 2 * t.half;
    const int a1 = a0 + 1;
    const float r0 = t.sV[vb + ((a0 < ALPH) ? a0 : 0)];   // clamped addr
    const float r1 = t.sV[vb + ((a1 < ALPH) ? a1 : 0)];
    va0[c] = (a0 < ALPH) ? r0 : 0.0f;                     // masked value
    va1[c] = (a1 < ALPH) ? r1 : 0.0f;
  }

#pragma unroll 2
  for (int li = 0; li < cnt; ++li) {
    const int l = l0 + li;
    const float* __restrict__ xl = t.xrow + l * ALPH;
    const int gbase = l * ALPH;
    __builtin_prefetch(xl + ALPH, 0, 3);

    float xb0[6], xb1[6];
#pragma unroll
    for (int c = 0; c < 6; ++c) {
      const int a0 = 4 * c + 2 * t.half;
      const int a1 = a0 + 1;
      // B operand: clamped address, UNMASKED value (A is 0 in padded-K slots,
      // and 0 * finite == 0, so garbage columns contribute nothing).
      xb0[c] = xl[(a0 < ALPH) ? a0 : 0];
      xb1[c] = xl[(a1 < ALPH) ? a1 : 0];
      // J values: needed by every lane each l; masked to 0 for padded K.
      const float j0r = t.sJ[gbase + ((a0 < ALPH) ? a0 : 0)];
      const float j1r = t.sJ[gbase + ((a1 < ALPH) ? a1 : 0)];
      const float j0 = (a0 < ALPH) ? j0r : 0.0f;
      const float j1 = (a1 < ALPH) ? j1r : 0.0f;
      jacc0 = fmaf(xb0[c], j0, jacc0);     // VALU, co-executes with WMMA
      jacc1 = fmaf(xb1[c], j1, jacc1);
    }
#pragma unroll
    for (int c = 0; c < 6; ++c) {
      v2f A;  A[0]  = (t.n == li) ? va0[c] : 0.0f;   // single active row
              A[1]  = (t.n == li) ? va1[c] : 0.0f;
      v2f Bm; Bm[0] = xb0[c]; Bm[1] = xb1[c];
      acc = wmma4(A, Bm, acc);
    }
  }
}

__global__ __launch_bounds__(256)
void cdr3_score_kernel(const float* __restrict__ x,
                       const float* __restrict__ vg,
                       const float* __restrict__ jg,
                       float* __restrict__ out) {
  __shared__ float sV[ROWE];
  __shared__ float sJ[ROWE];
  const int tid = threadIdx.x;
  for (int i = tid; i < ROWE; i += 256) { sV[i] = vg[i]; sJ[i] = jg[i]; }
  __syncthreads();

  const int lane = tid & 31;
  const int wave = tid >> 5;
  const int half = lane >> 4;
  const int n    = lane & 15;
  const long b0  = ((long)blockIdx.x * 8 + wave) * 16;

  TileCtx t;
  t.xrow = x + (b0 + n) * (long)ROWE;
  t.sV = sV; t.sJ = sJ; t.half = half; t.n = n;

  // acc0: positions 0..15 ; acc1: 16..31 ; acc2: rows 0-7 = pos 32..39
  v8f acc0 = {0,0,0,0,0,0,0,0};
  v8f acc1 = {0,0,0,0,0,0,0,0};
  v8f acc2 = {0,0,0,0,0,0,0,0};
  float jacc0 = 0.0f, jacc1 = 0.0f;

  tile_mma(t,  0, 16, acc0, jacc0, jacc1);
  tile_mma(t, 16, 16, acc1, jacc0, jacc1);
  tile_mma(t, 32,  8, acc2, jacc0, jacc1);

  // ---- epilogue: cumprod over positions, lane-local with half-carries ----
  // lane<16 holds positions {r}, {16+r}, {32+r}; lane>=16 holds {8+r}, {24+r}.
  float s0, p0, s1, p1, s2, p2;
  p0 = acc0[0]; s0 = p0;
#pragma unroll
  for (int i = 1; i < 8; ++i) { p0 *= acc0[i]; s0 += p0; }
  p1 = acc1[0]; s1 = p1;
#pragma unroll
  for (int i = 1; i < 8; ++i) { p1 *= acc1[i]; s1 += p1; }
  p2 = acc2[0]; s2 = p2;
#pragma unroll
  for (int i = 1; i < 8; ++i) { p2 *= acc2[i]; s2 += p2; }

  const float os0 = __shfl_xor(s0, 16, 32);
  const float op0 = __shfl_xor(p0, 16, 32);
  const float os1 = __shfl_xor(s1, 16, 32);
  const float op1 = __shfl_xor(p1, 16, 32);
  const float os2 = __shfl_xor(s2, 16, 32);

  // ordered groups: G0=pos0-7(h0), G1=8-15(h1), G2=16-23(h0), G3=24-31(h1), G4=32-39(h0)
  const float fs0 = half ? os0 : s0, fp0 = half ? op0 : p0;   // G0
  const float gs0 = half ? s0 : os0, gp0 = half ? p0 : op0;   // G1
  float S = fs0 + fp0 * gs0;
  float P = fp0 * gp0;
  const float fs1 = half ? os1 : s1, fp1 = half ? op1 : p1;   // G2
  const float gs1 = half ? s1 : os1, gp1 = half ? p1 : op1;   // G3
  S += P * (fs1 + fp1 * gs1);
  P *= fp1 * gp1;
  const float ls2 = half ? os2 : s2;                          // G4 (h0-sourced)
  S += P * ls2;

  // j_score: the two lane-halves hold disjoint amino partials of one batch elem
  const float jown = jacc0 + jacc1;
  const float jsum = jown + __shfl_xor(jown, 16, 32);

  if (half == 0) out[(b0 + n) * 2 + 0] = S;
  else           out[(b0 + n) * 2 + 1] = jsum;
}

extern "C" void kernel_launch(void* const* d_in, const int* in_sizes, int n_in,
                              void* d_out, int out_size, void* d_ws, size_t ws_size,
                              hipStream_t stream) {
  (void)n_in; (void)out_size; (void)d_ws; (void)ws_size;
  const float* x  = (const float*)d_in[0];
  const float* vg = (const float*)d_in[1];
  const float* jg = (const float*)d_in[2];
  float* out = (float*)d_out;

  const int B = in_sizes[0] / ROWE;        // 131072
  const int blocks = B / 128;              // 8 waves/block * 16 batch/wave
  cdr3_score_kernel<<<blocks, 256, 0, stream>>>(x, vg, jg, out);
}